// TripletLoss_18193481465868
// MI455X (gfx1250) — compile-verified
//
#include <hip/hip_runtime.h>
#include <hip/hip_bf16.h>
#include <math.h>

typedef __attribute__((ext_vector_type(2))) float v2f;
typedef __attribute__((ext_vector_type(8))) float v8f;

#define TM 32           // rows per block (M-tile: two 16-row WMMA tiles)
#define LDS_STRIDE 1028 // 1024 + 4 floats pad -> conflict-free A-fragment loads

// ---------------------------------------------------------------------------
// Kernel 1: per-row squared L2 norm. One wave per row.
// ---------------------------------------------------------------------------
__global__ __launch_bounds__(256)
void sqnorm_kernel(const float* __restrict__ X, float* __restrict__ sq, int K)
{
    const int row  = blockIdx.x * 8 + (threadIdx.x >> 5);
    const int lane = threadIdx.x & 31;
    const float4* p = (const float4*)(X + (size_t)row * K);
    float s = 0.f;
    for (int c = lane; c < K / 4; c += 32) {
        float4 v = p[c];
        s += v.x * v.x + v.y * v.y + v.z * v.z + v.w * v.w;
    }
    #pragma unroll
    for (int m = 1; m < 32; m <<= 1) s += __shfl_xor(s, m, 32);
    if (lane == 0) sq[row] = s;
}

// ---------------------------------------------------------------------------
// Kernel 2: fused GEMM (X X^T via v_wmma_f32_16x16x4_f32) + batch-hard mining.
// Block = 32 rows. Each of the 8 waves owns DISTINCT N-columns (no duplicate
// B fetches within a block) and computes BOTH 16-row M-halves for them:
// 8 accumulators / wave, 2 A-frag LDS loads + 4 B-frag global loads per K-step.
// ---------------------------------------------------------------------------
__global__ __launch_bounds__(256)
void triplet_mine_kernel(const float* __restrict__ X,
                         const int*   __restrict__ lbl,
                         const float* __restrict__ sq,
                         float* __restrict__ dap, float* __restrict__ dan,
                         int N, int K)
{
    __shared__ float As[TM][LDS_STRIDE];
    __shared__ float redAp[8][TM];
    __shared__ float redAn[8][TM];

    const int tid  = threadIdx.x;
    const int wave = tid >> 5;
    const int lane = tid & 31;
    const int m0   = blockIdx.x * TM;

    // --- stage A tile (TM x K) into LDS, coalesced float4 loads ---
    {
        const float4* src = (const float4*)(X + (size_t)m0 * K);
        const int kv = K / 4;
        for (int idx = tid; idx < TM * kv; idx += 256) {
            int r = idx / kv, c = idx % kv;
            float4 v = src[(size_t)r * kv + c];
            float* dst = &As[r][c * 4];
            dst[0] = v.x; dst[1] = v.y; dst[2] = v.z; dst[3] = v.w;
        }
    }
    __syncthreads();

    const int hl  = lane >> 4;   // half-lane
    const int l16 = lane & 15;

    // C-layout row metadata: for M-half s, lane covers rows s*16 + hl*8 + v
    float sqi[2][8]; int li[2][8];
    #pragma unroll
    for (int s = 0; s < 2; ++s) {
        const int rb = m0 + s * 16 + hl * 8;
        #pragma unroll
        for (int v = 0; v < 8; ++v) { sqi[s][v] = sq[rb + v]; li[s][v] = lbl[rb + v]; }
    }

    float ap[2][8], an[2][8];
    #pragma unroll
    for (int s = 0; s < 2; ++s)
        #pragma unroll
        for (int v = 0; v < 8; ++v) { ap[s][v] = -__builtin_inff(); an[s][v] = __builtin_inff(); }

    // A-fragment sources for the two M-halves: M = l16, K-pair by half-lane
    const float* aptr0 = &As[l16][2 * hl];
    const float* aptr1 = &As[16 + l16][2 * hl];

    const int NGRP = N / 64;     // groups of 4 N-tiles (128)
    for (int grp = wave; grp < NGRP; grp += 8) {
        const int nt0 = grp * 4;
        const float* b0 = X + (size_t)(nt0 * 16 + l16) * K + 2 * hl;
        const float* b1 = b0 + (size_t)16 * K;
        const float* b2 = b0 + (size_t)32 * K;
        const float* b3 = b0 + (size_t)48 * K;

        v8f c00 = {0.f}, c01 = {0.f}, c02 = {0.f}, c03 = {0.f};  // M-half 0
        v8f c10 = {0.f}, c11 = {0.f}, c12 = {0.f}, c13 = {0.f};  // M-half 1

        #pragma unroll 4
        for (int k = 0; k < K; k += 4) {
            v2f a0 = *(const v2f*)(aptr0 + k);
            v2f a1 = *(const v2f*)(aptr1 + k);
            v2f f0 = *(const v2f*)(b0 + k);
            v2f f1 = *(const v2f*)(b1 + k);
            v2f f2 = *(const v2f*)(b2 + k);
            v2f f3 = *(const v2f*)(b3 + k);
            c00 = __builtin_amdgcn_wmma_f32_16x16x4_f32(false, a0, false, f0, (short)0, c00, false, false);
            c10 = __builtin_amdgcn_wmma_f32_16x16x4_f32(false, a1, false, f0, (short)0, c10, false, false);
            c01 = __builtin_amdgcn_wmma_f32_16x16x4_f32(false, a0, false, f1, (short)0, c01, false, false);
            c11 = __builtin_amdgcn_wmma_f32_16x16x4_f32(false, a1, false, f1, (short)0, c11, false, false);
            c02 = __builtin_amdgcn_wmma_f32_16x16x4_f32(false, a0, false, f2, (short)0, c02, false, false);
            c12 = __builtin_amdgcn_wmma_f32_16x16x4_f32(false, a1, false, f2, (short)0, c12, false, false);
            c03 = __builtin_amdgcn_wmma_f32_16x16x4_f32(false, a0, false, f3, (short)0, c03, false, false);
            c13 = __builtin_amdgcn_wmma_f32_16x16x4_f32(false, a1, false, f3, (short)0, c13, false, false);
        }

        // fuse accumulators into running hard-pos max / hard-neg min
        const v8f* accs[2][4] = { { &c00, &c01, &c02, &c03 },
                                  { &c10, &c11, &c12, &c13 } };
        #pragma unroll
        for (int t = 0; t < 4; ++t) {
            const int   j   = (nt0 + t) * 16 + l16;   // column (lane-private)
            const float sqj = sq[j];
            const int   lj  = lbl[j];
            #pragma unroll
            for (int s = 0; s < 2; ++s) {
                const v8f& c = *accs[s][t];
                #pragma unroll
                for (int v = 0; v < 8; ++v) {
                    float d2  = sqi[s][v] + sqj - 2.0f * c[v];
                    bool same = (li[s][v] == lj);
                    ap[s][v] = (same  && d2 > ap[s][v]) ? d2 : ap[s][v];
                    an[s][v] = (!same && d2 < an[s][v]) ? d2 : an[s][v];
                }
            }
        }
    }

    // butterfly reduce across the 16 lanes of each half-wave; wave covers all
    // 32 block rows for its column subset
    #pragma unroll
    for (int s = 0; s < 2; ++s) {
        #pragma unroll
        for (int v = 0; v < 8; ++v) {
            float a_ = ap[s][v], n_ = an[s][v];
            #pragma unroll
            for (int m = 1; m < 16; m <<= 1) {
                a_ = fmaxf(a_, __shfl_xor(a_, m, 16));
                n_ = fminf(n_, __shfl_xor(n_, m, 16));
            }
            if (l16 == 0) {
                redAp[wave][s * 16 + hl * 8 + v] = a_;
                redAn[wave][s * 16 + hl * 8 + v] = n_;
            }
        }
    }
    __syncthreads();

    // combine the 8 waves; emit dist_ap / dist_an
    if (tid < TM) {
        float a_ = -__builtin_inff(), n_ = __builtin_inff();
        #pragma unroll
        for (int w = 0; w < 8; ++w) {
            a_ = fmaxf(a_, redAp[w][tid]);
            n_ = fminf(n_, redAn[w][tid]);
        }
        dap[m0 + tid] = sqrtf(fmaxf(a_, 1e-12f));
        dan[m0 + tid] = sqrtf(fmaxf(n_, 1e-12f));
    }
}

// ---------------------------------------------------------------------------
// Kernel 3: mean softplus(dist_ap - dist_an), single block.
// ---------------------------------------------------------------------------
__global__ __launch_bounds__(256)
void loss_kernel(const float* __restrict__ dap, const float* __restrict__ dan,
                 float* __restrict__ out, int n)
{
    __shared__ float red[256];
    float s = 0.f;
    for (int i = threadIdx.x; i < n; i += 256) {
        float x = dap[i] - dan[i];
        s += (x > 20.f) ? x : log1pf(expf(x));   // stable softplus
    }
    red[threadIdx.x] = s;
    __syncthreads();
    for (int o = 128; o > 0; o >>= 1) {
        if (threadIdx.x < o) red[threadIdx.x] += red[threadIdx.x + o];
        __syncthreads();
    }
    if (threadIdx.x == 0) out[0] = red[0] / (float)n;
}

// ---------------------------------------------------------------------------
extern "C" void kernel_launch(void* const* d_in, const int* in_sizes, int n_in,
                              void* d_out, int out_size, void* d_ws, size_t ws_size,
                              hipStream_t stream)
{
    const float* X   = (const float*)d_in[0];
    const int*   lbl = (const int*)d_in[1];
    const int N = in_sizes[1];              // 8192 rows
    const int K = in_sizes[0] / N;          // 1024 features

    float* sq  = (float*)d_ws;              // [N]
    float* dap = sq + N;                    // [N]
    float* dan = dap + N;                   // [N]

    sqnorm_kernel<<<N / 8, 256, 0, stream>>>(X, sq, K);
    triplet_mine_kernel<<<N / TM, 256, 0, stream>>>(X, lbl, sq, dap, dan, N, K);
    loss_kernel<<<1, 256, 0, stream>>>(dap, dan, (float*)d_out, N);
    (void)n_in; (void)out_size; (void)ws_size;
}